// GraphClassifier_34583076667495
// MI455X (gfx1250) — compile-verified
//
#include <hip/hip_runtime.h>

#define N_NODES    100000
#define N_EDGES    1600000
#define HID        64
#define N_CLASSES  10
#define NUM_GRAPHS 128

typedef float v2f __attribute__((ext_vector_type(2)));
typedef float v8f __attribute__((ext_vector_type(8)));

// ---------------------------------------------------------------- zero fill
__global__ void zero_f32(float* __restrict__ p, int n) {
    int i = blockIdx.x * blockDim.x + threadIdx.x;
    if (i < n) p[i] = 0.0f;
}

// ------------------------------------------- edge scatter: agg[dst] += feat[src]
// one thread per (edge, channel); 64 consecutive lanes share one edge so the
// feature reads and atomic adds are fully coalesced 256B transactions.
__global__ void scatter_add64(const float* __restrict__ feat,
                              const int*  __restrict__ src,
                              const int*  __restrict__ dst,
                              float* __restrict__ agg) {
    unsigned int tid = blockIdx.x * blockDim.x + threadIdx.x;
    if (tid >= (unsigned int)N_EDGES * 64u) return;
    unsigned int e  = tid >> 6;
    unsigned int ch = tid & 63u;
    int s = src[e];
    int d = dst[e];
    atomicAdd(&agg[(size_t)d * HID + ch], feat[(size_t)s * HID + ch]);
}

// ------------------------------------------------- fused GraphConv GEMM (WMMA)
// out[16,64] = relu(agg[16,64] @ w_rel + xin[16,64] @ w_root + bias)
// One wave per 16-node row tile. f32 WMMA 16x16x4:
//   A frag (16x4):  lane L holds row M=L%16, a[v] = A[M][kbase + v + 2*(L>>4)]
//   B frag (4x16):  lane L holds col N=L%16, b[v] = B[kbase + v + 2*(L>>4)][N]
//   C/D (16x16):    lane L, vgpr v -> M = v + 8*(L>>4), N = L%16
__global__ __launch_bounds__(128) void graphconv_gemm(
        const float* __restrict__ agg,    // [N,64] neighbor sums
        const float* __restrict__ xin,    // [N,64] root features
        const float* __restrict__ w_rel,  // [64,64]
        const float* __restrict__ w_root, // [64,64]
        const float* __restrict__ bias,   // [64]
        float* __restrict__ out)          // [N,64]
{
    const int ntiles = N_NODES / 16;
    int tile = blockIdx.x * 4 + (threadIdx.x >> 5);
    if (tile >= ntiles) return;                 // whole-wave guard: EXEC all 1s
    int lane  = threadIdx.x & 31;
    int row   = lane & 15;
    int khalf = lane >> 4;                      // 0: K pair base, 1: +2
    size_t base = (size_t)tile * 16;

    // Preload both A tiles (16x64 each) into registers: 16 K-steps of float2.
    v2f a_agg[16], a_x[16];
#pragma unroll
    for (int kk = 0; kk < 16; ++kk) {
        size_t off = (base + row) * HID + 4 * kk + 2 * khalf;
        a_agg[kk] = *(const v2f*)(agg + off);
        a_x[kk]   = *(const v2f*)(xin + off);
    }

#pragma unroll
    for (int jt = 0; jt < 4; ++jt) {            // 4 output column tiles of 16
        int col = 16 * jt + row;
        float bv = bias[col];
        v8f acc = {bv, bv, bv, bv, bv, bv, bv, bv};

        v2f brel[16], broot[16];
#pragma unroll
        for (int kk = 0; kk < 16; ++kk) {       // B frags (weights stay in L2)
            int kb = 4 * kk + 2 * khalf;
            brel[kk].x  = w_rel [(kb    ) * HID + col];
            brel[kk].y  = w_rel [(kb + 1) * HID + col];
            broot[kk].x = w_root[(kb    ) * HID + col];
            broot[kk].y = w_root[(kb + 1) * HID + col];
        }
#pragma unroll
        for (int kk = 0; kk < 16; ++kk)
            acc = __builtin_amdgcn_wmma_f32_16x16x4_f32(
                false, a_agg[kk], false, brel[kk], (short)0, acc, false, false);
#pragma unroll
        for (int kk = 0; kk < 16; ++kk)
            acc = __builtin_amdgcn_wmma_f32_16x16x4_f32(
                false, a_x[kk], false, broot[kk], (short)0, acc, false, false);

#pragma unroll
        for (int v = 0; v < 8; ++v) {           // fused ReLU + store
            float val = acc[v];
            val = fmaxf(val, 0.0f);
            out[(base + v + 8 * khalf) * HID + col] = val;
        }
    }
}

// ------------------------------------------------------ global mean-pool (sum)
__global__ void pool_kernel(const float* __restrict__ h,
                            const int*  __restrict__ batch,
                            float* __restrict__ pooled,   // [G,64]
                            float* __restrict__ counts) { // [G]
    int tid = blockIdx.x * blockDim.x + threadIdx.x;
    if (tid >= N_NODES * HID) return;
    int node = tid >> 6;
    int ch   = tid & 63;
    int g = batch[node];
    atomicAdd(&pooled[(size_t)g * HID + ch], h[tid]);
    if (ch == 0) atomicAdd(&counts[g], 1.0f);
}

// ------------------------------------------------------------------ head GEMM
__global__ void final_kernel(const float* __restrict__ pooled,
                             const float* __restrict__ counts,
                             const float* __restrict__ w_out, // [64,10]
                             const float* __restrict__ b_out, // [10]
                             float* __restrict__ out) {       // [G,10]
    int tid = blockIdx.x * blockDim.x + threadIdx.x;
    if (tid >= NUM_GRAPHS * N_CLASSES) return;
    int g = tid / N_CLASSES;
    int c = tid % N_CLASSES;
    float inv = 1.0f / fmaxf(counts[g], 1.0f);
    float s = 0.0f;
#pragma unroll
    for (int k = 0; k < HID; ++k)
        s += pooled[(size_t)g * HID + k] * w_out[k * N_CLASSES + c];
    out[tid] = s * inv + b_out[c];
}

// ----------------------------------------------------------------------------
extern "C" void kernel_launch(void* const* d_in, const int* in_sizes, int n_in,
                              void* d_out, int out_size, void* d_ws, size_t ws_size,
                              hipStream_t stream) {
    const float* x       = (const float*)d_in[0];
    const int*   ei      = (const int*)  d_in[1];   // [2, N_EDGES]
    const int*   batch   = (const int*)  d_in[2];   // [N]
    const float* w1_rel  = (const float*)d_in[3];
    const float* w1_root = (const float*)d_in[4];
    const float* b1      = (const float*)d_in[5];
    const float* w2_rel  = (const float*)d_in[6];
    const float* w2_root = (const float*)d_in[7];
    const float* b2      = (const float*)d_in[8];
    const float* w_out   = (const float*)d_in[9];
    const float* b_out   = (const float*)d_in[10];
    float* out = (float*)d_out;

    // workspace layout (all 256B-aligned offsets)
    char*  ws     = (char*)d_ws;
    float* bufA   = (float*)(ws);                                    // agg [N,64]
    float* bufH   = (float*)(ws + (size_t)N_NODES * HID * 4);        // h   [N,64]
    float* pooled = (float*)(ws + (size_t)2 * N_NODES * HID * 4);    // [G,64]
    float* counts = pooled + NUM_GRAPHS * HID;                       // [G]

    const int* src = ei;
    const int* dst = ei + N_EDGES;

    const int nfeat    = N_NODES * HID;        // 6.4M
    const int nedge_el = N_EDGES * 64;         // 102.4M
    const int gemm_blocks = (N_NODES / 16 + 3) / 4;

    // ---- layer 1: agg = scatter(x); h1 = relu(agg@W1r + x@W1o + b1)
    zero_f32<<<(nfeat + 255) / 256, 256, 0, stream>>>(bufA, nfeat);
    scatter_add64<<<(nedge_el + 255) / 256, 256, 0, stream>>>(x, src, dst, bufA);
    graphconv_gemm<<<gemm_blocks, 128, 0, stream>>>(bufA, x, w1_rel, w1_root, b1, bufH);

    // ---- layer 2: agg = scatter(h1); h2 = relu(agg@W2r + h1@W2o + b2) (in place)
    zero_f32<<<(nfeat + 255) / 256, 256, 0, stream>>>(bufA, nfeat);
    scatter_add64<<<(nedge_el + 255) / 256, 256, 0, stream>>>(bufH, src, dst, bufA);
    graphconv_gemm<<<gemm_blocks, 128, 0, stream>>>(bufA, bufH, w2_rel, w2_root, b2, bufH);

    // ---- mean pool + head
    int npool = NUM_GRAPHS * HID + NUM_GRAPHS;  // pooled + counts are contiguous
    zero_f32<<<(npool + 255) / 256, 256, 0, stream>>>(pooled, npool);
    pool_kernel<<<(nfeat + 255) / 256, 256, 0, stream>>>(bufH, batch, pooled, counts);
    final_kernel<<<(NUM_GRAPHS * N_CLASSES + 255) / 256, 256, 0, stream>>>(
        pooled, counts, w_out, b_out, out);
}